// GeneLink_5351529251386
// MI455X (gfx1250) — compile-verified
//
#include <hip/hip_runtime.h>
#include <math.h>

typedef float v2f __attribute__((ext_vector_type(2)));
typedef float v4f __attribute__((ext_vector_type(4)));
typedef float v8f __attribute__((ext_vector_type(8)));

#define N_NODES 20000
#define HC      256          // H*C (same both layers)
#define NHEADS  4
#define CHEAD   64
#define E_RAND  640000
#define E_TOT   (E_RAND + N_NODES)   // 660000 incl. self loops
#define NEG_SLOPE 0.2f

// ---------------------------------------------------------------------------
// GEMM: X[M,K] @ W[K,256] -> OUT[M,256] using V_WMMA_F32_16X16X4_F32.
// Block = 256 threads = 8 waves. Block owns 16 rows; wave w owns cols
// [32w, 32w+32) as two 16x16 accumulator tiles. A staged via LDS (stride 68
// floats -> conflict-free float2 reads), B served from L0/L2 (<=512KB, reused
// by all 1250 blocks).
// ---------------------------------------------------------------------------
template <int K>
__global__ __launch_bounds__(256) void gemm_wmma_kernel(
    const float* __restrict__ X, const float* __restrict__ W,
    float* __restrict__ OUT)
{
    __shared__ float lds[16 * 68];
    const int tid  = threadIdx.x;
    const int lane = tid & 31;
    const int half = lane >> 4;   // 0: lanes 0-15, 1: lanes 16-31
    const int l15  = lane & 15;
    const int wave = tid >> 5;
    const int m0   = blockIdx.x * 16;
    const int colb = wave * 32;

    v8f acc0 = {};
    v8f acc1 = {};

    const int lr  = tid >> 4;        // loader row 0..15
    const int lk4 = (tid & 15) * 4;  // loader k offset 0..60

    for (int kc = 0; kc < K; kc += 64) {
        // coalesced stage of A chunk [16 x 64] into LDS
        v4f av = *(const v4f*)(X + (m0 + lr) * K + kc + lk4);
        *(v4f*)(&lds[lr * 68 + lk4]) = av;
        __syncthreads();
#pragma unroll
        for (int kk = 0; kk < 64; kk += 4) {
            // A frag: VGPR v holds A[row][kk + 2*half + v]
            v2f a = *(const v2f*)(&lds[l15 * 68 + kk + 2 * half]);
            const int krow = kc + kk + 2 * half;
            const int c0   = colb + l15;
            v2f b0, b1;
            b0.x = W[(krow    ) * HC + c0];
            b0.y = W[(krow + 1) * HC + c0];
            b1.x = W[(krow    ) * HC + c0 + 16];
            b1.y = W[(krow + 1) * HC + c0 + 16];
            acc0 = __builtin_amdgcn_wmma_f32_16x16x4_f32(
                false, a, false, b0, (short)0, acc0, false, false);
            acc1 = __builtin_amdgcn_wmma_f32_16x16x4_f32(
                false, a, false, b1, (short)0, acc1, false, false);
        }
        __syncthreads();
    }
    // C/D layout: VGPR v -> row (v + 8*half), col = lane&15
#pragma unroll
    for (int v = 0; v < 8; ++v) {
        const int row = m0 + v + 8 * half;
        OUT[row * HC + colb + l15]      = acc0[v];
        OUT[row * HC + colb + l15 + 16] = acc1[v];
    }
}

// ---------------------------------------------------------------------------
// Per-node attention logits: asrc[n,h] = <h[n,h,:], att_src[h,:]>, same adst.
// ---------------------------------------------------------------------------
__global__ void node_att_kernel(const float* __restrict__ h,
                                const float* __restrict__ a_src,
                                const float* __restrict__ a_dst,
                                float* __restrict__ asrc,
                                float* __restrict__ adst)
{
    const int i = blockIdx.x * blockDim.x + threadIdx.x;  // (n, head)
    if (i >= N_NODES * NHEADS) return;
    const int n  = i >> 2;
    const int hd = i & 3;
    const float* hp = h + n * HC + hd * CHEAD;
    const float* sp = a_src + hd * CHEAD;
    const float* dp = a_dst + hd * CHEAD;
    float s = 0.f, d = 0.f;
#pragma unroll 8
    for (int c = 0; c < CHEAD; ++c) {
        const float v = hp[c];
        s += v * sp[c];
        d += v * dp[c];
    }
    asrc[i] = s;
    adst[i] = d;
}

__global__ void init_l1_kernel(float* __restrict__ agg,
                               float* __restrict__ emax,
                               float* __restrict__ denom)
{
    const int i = blockIdx.x * blockDim.x + threadIdx.x;
    if (i < N_NODES * HC) agg[i] = 0.f;
    if (i < N_NODES * NHEADS) { emax[i] = -INFINITY; denom[i] = 0.f; }
}

__global__ void init_l2_kernel(float* __restrict__ out,
                               const float* __restrict__ b2,
                               float* __restrict__ emax,
                               float* __restrict__ denom)
{
    const int i = blockIdx.x * blockDim.x + threadIdx.x;
    if (i < N_NODES * HC) out[i] = b2[i & (HC - 1)];  // scatter adds on top
    if (i < N_NODES * NHEADS) { emax[i] = -INFINITY; denom[i] = 0.f; }
}

// monotone int-bits float atomic max
__device__ __forceinline__ void atomicMaxF(float* a, float v)
{
    if (v >= 0.f) atomicMax((int*)a, __float_as_int(v));
    else          atomicMin((unsigned int*)a, __float_as_uint(v));
}

__global__ void edge_logit_kernel(const long long* __restrict__ src,
                                  const long long* __restrict__ dst,
                                  const float* __restrict__ asrc,
                                  const float* __restrict__ adst,
                                  float* __restrict__ ebuf,
                                  float* __restrict__ emax)
{
    const int i = blockIdx.x * blockDim.x + threadIdx.x;  // (e, head)
    if (i >= E_TOT * NHEADS) return;
    const int e  = i >> 2;
    const int hd = i & 3;
    int s, d;
    if (e < E_RAND) { s = (int)src[e]; d = (int)dst[e]; }
    else            { s = d = e - E_RAND; }               // self loop
    float v = asrc[s * NHEADS + hd] + adst[d * NHEADS + hd];
    v = (v > 0.f) ? v : v * NEG_SLOPE;                    // leaky relu
    ebuf[i] = v;
    atomicMaxF(&emax[d * NHEADS + hd], v);
}

__global__ void edge_exp_kernel(const long long* __restrict__ dst,
                                const float* __restrict__ emax,
                                float* __restrict__ ebuf,
                                float* __restrict__ denom)
{
    const int i = blockIdx.x * blockDim.x + threadIdx.x;
    if (i >= E_TOT * NHEADS) return;
    const int e  = i >> 2;
    const int hd = i & 3;
    const int d  = (e < E_RAND) ? (int)dst[e] : (e - E_RAND);
    const float ee = expf(ebuf[i] - emax[d * NHEADS + hd]);
    ebuf[i] = ee;
    atomicAdd(&denom[d * NHEADS + hd], ee);
}

// ---------------------------------------------------------------------------
// One wave per edge: lane handles 8 channels (c = lane + 32j), coalesced 128B
// gathers of h[src] (L2-resident, 20.5MB) and f32 atomic adds into agg[dst].
// ---------------------------------------------------------------------------
__global__ __launch_bounds__(256) void scatter_kernel(
    const long long* __restrict__ src, const long long* __restrict__ dst,
    const float* __restrict__ h, const float* __restrict__ ebuf,
    const float* __restrict__ denom, float* __restrict__ agg)
{
    const int e    = blockIdx.x * (blockDim.x >> 5) + (threadIdx.x >> 5);
    const int lane = threadIdx.x & 31;
    if (e >= E_TOT) return;
    int s, d;
    if (e < E_RAND) { s = (int)src[e]; d = (int)dst[e]; }
    else            { s = d = e - E_RAND; }
    float alpha[NHEADS];
#pragma unroll
    for (int hd = 0; hd < NHEADS; ++hd)
        alpha[hd] = ebuf[e * NHEADS + hd] / denom[d * NHEADS + hd];
    const float* hp = h + s * HC;
    float* op = agg + d * HC;
#pragma unroll
    for (int j = 0; j < 8; ++j) {
        const int c = lane + 32 * j;         // head = c/64 = j>>1
        atomicAdd(&op[c], hp[c] * alpha[j >> 1]);
    }
}

__global__ void bias_elu_kernel(const float* __restrict__ agg,
                                const float* __restrict__ b,
                                float* __restrict__ out)
{
    const int i = blockIdx.x * blockDim.x + threadIdx.x;
    if (i >= N_NODES * HC) return;
    const float v = agg[i] + b[i & (HC - 1)];
    out[i] = (v > 0.f) ? v : (expf(v) - 1.f);   // ELU alpha=1
}

// ---------------------------------------------------------------------------
extern "C" void kernel_launch(void* const* d_in, const int* in_sizes, int n_in,
                              void* d_out, int out_size, void* d_ws, size_t ws_size,
                              hipStream_t stream)
{
    const float*     x   = (const float*)d_in[0];
    const long long* ei  = (const long long*)d_in[1];   // int64 [2, 640000]
    const float*     W1  = (const float*)d_in[2];
    const float*     as1 = (const float*)d_in[3];
    const float*     ad1 = (const float*)d_in[4];
    const float*     b1  = (const float*)d_in[5];
    const float*     W2  = (const float*)d_in[6];
    const float*     as2 = (const float*)d_in[7];
    const float*     ad2 = (const float*)d_in[8];
    const float*     b2  = (const float*)d_in[9];
    float* out = (float*)d_out;

    const long long* src = ei;
    const long long* dst = ei + E_RAND;

    // workspace layout (~53 MB)
    float* B0    = (float*)d_ws;                 // h matrix      [N, 256]
    float* B1    = B0 + N_NODES * HC;            // agg / h2      [N, 256]
    float* EB    = B1 + N_NODES * HC;            // edge buf      [E_TOT, 4]
    float* asrc  = EB + E_TOT * NHEADS;          // [N, 4]
    float* adst  = asrc + N_NODES * NHEADS;
    float* emax  = adst + N_NODES * NHEADS;
    float* denom = emax + N_NODES * NHEADS;

    const dim3 blk(256);
    const int nb_nhc  = (N_NODES * HC + 255) / 256;
    const int nb_nh   = (N_NODES * NHEADS + 255) / 256;
    const int nb_eh   = (E_TOT * NHEADS + 255) / 256;
    const int nb_edge = (E_TOT + 7) / 8;         // 8 waves per block

    // ---- layer 1 ----
    gemm_wmma_kernel<512><<<N_NODES / 16, blk, 0, stream>>>(x, W1, B0);
    node_att_kernel<<<nb_nh, blk, 0, stream>>>(B0, as1, ad1, asrc, adst);
    init_l1_kernel<<<nb_nhc, blk, 0, stream>>>(B1, emax, denom);
    edge_logit_kernel<<<nb_eh, blk, 0, stream>>>(src, dst, asrc, adst, EB, emax);
    edge_exp_kernel<<<nb_eh, blk, 0, stream>>>(dst, emax, EB, denom);
    scatter_kernel<<<nb_edge, blk, 0, stream>>>(src, dst, B0, EB, denom, B1);
    bias_elu_kernel<<<nb_nhc, blk, 0, stream>>>(B1, b1, B0);   // act -> B0

    // ---- layer 2 ----
    gemm_wmma_kernel<256><<<N_NODES / 16, blk, 0, stream>>>(B0, W2, B1);
    node_att_kernel<<<nb_nh, blk, 0, stream>>>(B1, as2, ad2, asrc, adst);
    init_l2_kernel<<<nb_nhc, blk, 0, stream>>>(out, b2, emax, denom);
    edge_logit_kernel<<<nb_eh, blk, 0, stream>>>(src, dst, asrc, adst, EB, emax);
    edge_exp_kernel<<<nb_eh, blk, 0, stream>>>(dst, emax, EB, denom);
    scatter_kernel<<<nb_edge, blk, 0, stream>>>(src, dst, B1, EB, denom, out);
}